// InfoNCELoss_80212809220156
// MI455X (gfx1250) — compile-verified
//
#include <hip/hip_runtime.h>
#include <math.h>

typedef __attribute__((ext_vector_type(2))) float v2f;
typedef __attribute__((ext_vector_type(8))) float v8f;

#define D       512
#define BM      64          // rows per workgroup
#define BN      64          // cols per column-iteration
#define STRIDE  516         // 512 + 4 pad -> bank-conflict-free (516 % 64 == 4)
#define CSPLIT  4           // column-chunk split for occupancy

__device__ __forceinline__ const float* row_ptr(const float* zi, const float* zj,
                                                int r, int B) {
  return (r < B) ? (zi + (size_t)r * D) : (zj + (size_t)(r - B) * D);
}

// ---------------- Kernel 1: inverse row norms + zero accumulators ----------------
__global__ __launch_bounds__(256) void norm_zero_kernel(
    const float* __restrict__ zi, const float* __restrict__ zj,
    float* __restrict__ invn, float* __restrict__ s1, float* __restrict__ s2,
    float* __restrict__ out, int B) {
  const int w    = threadIdx.x >> 5;   // wave in block (8 waves)
  const int lane = threadIdx.x & 31;
  const int row  = blockIdx.x * 8 + w; // one wave per row
  const float4* p4 = (const float4*)row_ptr(zi, zj, row, B);
  float s = 0.f;
#pragma unroll
  for (int i = 0; i < 4; ++i) {        // 32 lanes * 4 float4 = 512 floats
    float4 v = p4[lane + 32 * i];
    s += v.x * v.x + v.y * v.y + v.z * v.z + v.w * v.w;
  }
#pragma unroll
  for (int m = 16; m >= 1; m >>= 1) s += __shfl_xor(s, m, 32);
  if (lane == 0) {
    invn[row] = 1.0f / fmaxf(sqrtf(s), 1e-12f);
    s1[row] = 0.f;
    s2[row] = 0.f;
  }
  if (blockIdx.x == 0 && threadIdx.x == 0) out[0] = 0.f;
}

// ---------------- Kernels 2/3: fused WMMA GEMM + exp-sum passes ----------------
// PASS 0: S1[r] = sum_j exp(sim[r][j])  (diag excluded), capture e_label
// PASS 1: S2[r] = sum_j exp(exp(sim[r][j]) / S1[r])  (diag -> exp(0)=1 automatic)
template <int PASS>
__global__ __launch_bounds__(256) void gemm_pass_kernel(
    const float* __restrict__ zi, const float* __restrict__ zj,
    const float* __restrict__ invn,
    float* __restrict__ s1, float* __restrict__ s2,
    float* __restrict__ elab, int B) {
  __shared__ float ldsA[BM * STRIDE];   // 132 KB
  __shared__ float ldsB[BN * STRIDE];   // 132 KB
  __shared__ float ldsRowS[BM];
  __shared__ float ldsInv[BM];

  const int N       = 2 * B;
  const int tid     = threadIdx.x;
  const int rowBase = blockIdx.x * BM;
  const int chunk   = N / CSPLIT;
  const int cStart  = blockIdx.y * chunk;
  const int cEnd    = cStart + chunk;

  if (tid < BM) {
    ldsRowS[tid] = 0.f;
    if (PASS == 1) ldsInv[tid] = 1.0f / s1[rowBase + tid];
  }

  // Stage A panel (64 rows x 512), normalized on the fly. 16B LDS stores.
  for (int f = tid; f < BM * (D / 4); f += 256) {
    const int r  = f >> 7;             // 128 float4 per row
    const int c4 = f & 127;
    const int gr = rowBase + r;
    const float sc = invn[gr];
    float4 v = ((const float4*)row_ptr(zi, zj, gr, B))[c4];
    float4 o; o.x = v.x * sc; o.y = v.y * sc; o.z = v.z * sc; o.w = v.w * sc;
    *(float4*)&ldsA[r * STRIDE + c4 * 4] = o;
  }

  const int lane = tid & 31;
  const int w    = tid >> 5;
  const int lm   = lane & 15;          // N (or M) index within 16
  const int hi   = lane >> 4;          // half-wave select
  const int kb   = hi * 2;             // K base for A/B operands
  const int m0   = (w & 3) * 16;       // 4 tile-rows
  const int n0a  = (w >> 2) * 16;      // 2 tile-cols per wave ...
  const int n0b  = n0a + 32;           // ... covering all 4 tile-cols

  for (int cb = cStart; cb < cEnd; cb += BN) {
    __syncthreads();                   // previous compute done before overwrite
    // Stage B panel (64 cols-as-rows x 512), normalized on the fly
    for (int f = tid; f < BN * (D / 4); f += 256) {
      const int r  = f >> 7;
      const int c4 = f & 127;
      const int gc = cb + r;
      const float sc = invn[gc];
      float4 v = ((const float4*)row_ptr(zi, zj, gc, B))[c4];
      float4 o; o.x = v.x * sc; o.y = v.y * sc; o.z = v.z * sc; o.w = v.w * sc;
      *(float4*)&ldsB[r * STRIDE + c4 * 4] = o;
    }
    __syncthreads();

    // Two independent 16x16 accumulator chains per wave, shared A operand.
    v8f acc0 = {};
    v8f acc1 = {};
    const float* pa  = &ldsA[(m0  + lm) * STRIDE + kb];
    const float* pb0 = &ldsB[(n0a + lm) * STRIDE + kb];
    const float* pb1 = &ldsB[(n0b + lm) * STRIDE + kb];
#pragma unroll 8
    for (int kk = 0; kk < D; kk += 4) {
      v2f a  = *(const v2f*)(pa  + kk);
      v2f b0 = *(const v2f*)(pb0 + kk);
      v2f b1 = *(const v2f*)(pb1 + kk);
      acc0 = __builtin_amdgcn_wmma_f32_16x16x4_f32(false, a, false, b0,
                                                   (short)0, acc0, false, false);
      acc1 = __builtin_amdgcn_wmma_f32_16x16x4_f32(false, a, false, b1,
                                                   (short)0, acc1, false, false);
    }

    // Epilogue: exp, diag mask, label capture, 16-lane row reduce, LDS atomics
#pragma unroll
    for (int i = 0; i < 8; ++i) {
      const int lrow = m0 + i + 8 * hi;     // C layout: vgpr i -> M = i + 8*hi
      const int grow = rowBase + lrow;
      // ---- tile (m0, n0a)
      {
        const int gcol = cb + n0a + lm;
        const float sv = acc0[i];
        float e = (gcol == grow) ? 0.f : __expf(sv);
        if (PASS == 0) {
          const int part = (grow < B) ? grow + B : grow - B;
          if (gcol == part) elab[grow] = e;
        }
        float c = (PASS == 0) ? e : __expf(e * ldsInv[lrow]);
        c += __shfl_xor(c, 1, 32);
        c += __shfl_xor(c, 2, 32);
        c += __shfl_xor(c, 4, 32);
        c += __shfl_xor(c, 8, 32);
        if (lm == 0) atomicAdd(&ldsRowS[lrow], c);
      }
      // ---- tile (m0, n0b)
      {
        const int gcol = cb + n0b + lm;
        const float sv = acc1[i];
        float e = (gcol == grow) ? 0.f : __expf(sv);
        if (PASS == 0) {
          const int part = (grow < B) ? grow + B : grow - B;
          if (gcol == part) elab[grow] = e;
        }
        float c = (PASS == 0) ? e : __expf(e * ldsInv[lrow]);
        c += __shfl_xor(c, 1, 32);
        c += __shfl_xor(c, 2, 32);
        c += __shfl_xor(c, 4, 32);
        c += __shfl_xor(c, 8, 32);
        if (lm == 0) atomicAdd(&ldsRowS[lrow], c);
      }
    }
  }

  __syncthreads();
  if (tid < BM) {
    if (PASS == 0) atomicAdd(&s1[rowBase + tid], ldsRowS[tid]);
    else           atomicAdd(&s2[rowBase + tid], ldsRowS[tid]);
  }
}

// ---------------- Kernel 4: nll_r = log(S2) - ELab/S1, mean into out ----------------
__global__ __launch_bounds__(256) void finalize_kernel(
    const float* __restrict__ s1, const float* __restrict__ s2,
    const float* __restrict__ elab, float* __restrict__ out, int N) {
  const int r = blockIdx.x * 256 + threadIdx.x;
  const float nll = logf(s2[r]) - elab[r] / s1[r];
  atomicAdd(out, nll * (1.0f / (float)N));
}

// ---------------- host launch ----------------
extern "C" void kernel_launch(void* const* d_in, const int* in_sizes, int n_in,
                              void* d_out, int out_size, void* d_ws, size_t ws_size,
                              hipStream_t stream) {
  const float* zi = (const float*)d_in[0];
  const float* zj = (const float*)d_in[1];
  float* out = (float*)d_out;
  const int B = in_sizes[0] / D;   // 4096
  const int N = 2 * B;             // 8192

  float* ws   = (float*)d_ws;      // 4*N floats = 128 KB of scratch
  float* invn = ws;                // [0,   N)
  float* s1   = ws + N;            // [N,  2N)
  float* elab = ws + 2 * N;        // [2N, 3N)
  float* s2   = ws + 3 * N;        // [3N, 4N)

  norm_zero_kernel<<<N / 8, 256, 0, stream>>>(zi, zj, invn, s1, s2, out, B);

  dim3 grid(N / BM, CSPLIT);
  gemm_pass_kernel<0><<<grid, 256, 0, stream>>>(zi, zj, invn, s1, s2, elab, B);
  gemm_pass_kernel<1><<<grid, 256, 0, stream>>>(zi, zj, invn, s1, s2, elab, B);

  finalize_kernel<<<N / 256, 256, 0, stream>>>(s1, s2, elab, out, N);
}